// HMDIMUModel_8332236554437
// MI455X (gfx1250) — compile-verified
//
#include <hip/hip_runtime.h>
#include <hip/hip_bf16.h>
#include <stdint.h>

#define NJ    22
#define TPB   128
#define ROWF  132   // floats per sample (22 joints * 6)
#define ROWF4 33    // float4 per sample (528 bytes, 16B aligned)

// ---- CDNA5 async global<->LDS path (guarded; falls back to sync copy) ----
#if __has_builtin(__builtin_amdgcn_global_load_async_to_lds_b128) && \
    __has_builtin(__builtin_amdgcn_global_store_async_from_lds_b128)
#define HAVE_ASYNC_LDS 1
#endif

typedef int v4i __attribute__((ext_vector_type(4)));
typedef __attribute__((address_space(1))) v4i* gv4i_p;  // global int4*
typedef __attribute__((address_space(3))) v4i* lv4i_p;  // LDS int4*

__device__ __forceinline__ void wait_async0() {
#if __has_builtin(__builtin_amdgcn_s_wait_asynccnt)
    __builtin_amdgcn_s_wait_asynccnt(0);
#else
    asm volatile("s_wait_asynccnt 0" ::: "memory");
#endif
}

__global__ __launch_bounds__(TPB) void pose6d_fk_kernel(const float* __restrict__ in,
                                                        float* __restrict__ out,
                                                        int nSamples) {
    __shared__ float lds[TPB * ROWF];   // 67,584 bytes
    const int tid = threadIdx.x;
    const int s0  = blockIdx.x * TPB;
    int ns = nSamples - s0;
    if (ns > TPB) ns = TPB;
    const int nf4 = ns * ROWF4;

    const float4* __restrict__ g4in  = (const float4*)in  + (size_t)s0 * ROWF4;
    float4* __restrict__       g4out = (float4*)out       + (size_t)s0 * ROWF4;
    float4* l4 = (float4*)lds;

    // ---------- Phase 1: coalesced global -> LDS (async DMA path) ----------
#ifdef HAVE_ASYNC_LDS
    for (int i = tid; i < nf4; i += TPB) {
        __builtin_amdgcn_global_load_async_to_lds_b128(
            (gv4i_p)(uintptr_t)(g4in + i),
            (lv4i_p)(uint32_t)(uintptr_t)(l4 + i),
            0, 0);
    }
    wait_async0();
#else
    for (int i = tid; i < nf4; i += TPB) l4[i] = g4in[i];
#endif
    __syncthreads();

    // ---------- Phase 2: per-sample 6D->R + kinematic chain, in place ----------
    if (tid < ns) {
        float* row = &lds[tid * ROWF];
        constexpr int PAR[NJ] = {0,0,0,0,1,2,3,4,5,6,7,8,9,9,9,12,13,14,16,17,18,19};
        float G[NJ][9];   // fully unrolled -> constant indices -> registers
#pragma unroll
        for (int j = 0; j < NJ; ++j) {
            const float a1x = row[j*6+0], a1y = row[j*6+1], a1z = row[j*6+2];
            const float a2x = row[j*6+3], a2y = row[j*6+4], a2z = row[j*6+5];
            // b1 = a1 / max(|a1|, eps)
            const float n1 = sqrtf(a1x*a1x + a1y*a1y + a1z*a1z);
            const float i1 = 1.0f / fmaxf(n1, 1e-12f);
            const float b1x = a1x*i1, b1y = a1y*i1, b1z = a1z*i1;
            // a2p = a2 - (b1.a2) b1 ; b2 = a2p / max(|a2p|, eps)
            const float d  = b1x*a2x + b1y*a2y + b1z*a2z;
            const float ux = a2x - d*b1x, uy = a2y - d*b1y, uz = a2z - d*b1z;
            const float n2 = sqrtf(ux*ux + uy*uy + uz*uz);
            const float i2 = 1.0f / fmaxf(n2, 1e-12f);
            const float b2x = ux*i2, b2y = uy*i2, b2z = uz*i2;
            // b3 = b1 x b2
            const float b3x = b1y*b2z - b1z*b2y;
            const float b3y = b1z*b2x - b1x*b2z;
            const float b3z = b1x*b2y - b1y*b2x;
            const float L[9] = {b1x,b1y,b1z, b2x,b2y,b2z, b3x,b3y,b3z};
            float* g = G[j];
            if (j == 0) {
#pragma unroll
                for (int q = 0; q < 9; ++q) g[q] = L[q];
            } else {
                const float* p = G[PAR[j]];   // PAR[j] is a compile-time constant
#pragma unroll
                for (int r = 0; r < 3; ++r) {
#pragma unroll
                    for (int c = 0; c < 3; ++c) {
                        g[r*3+c] = p[r*3+0]*L[0*3+c]
                                 + p[r*3+1]*L[1*3+c]
                                 + p[r*3+2]*L[2*3+c];
                    }
                }
            }
            // output rows 0,1 of R_global overwrite this joint's input slot
#pragma unroll
            for (int q = 0; q < 6; ++q) row[j*6+q] = g[q];
        }
    }
    __syncthreads();

    // ---------- Phase 3: coalesced LDS -> global (async DMA path) ----------
#ifdef HAVE_ASYNC_LDS
    for (int i = tid; i < nf4; i += TPB) {
        __builtin_amdgcn_global_store_async_from_lds_b128(
            (gv4i_p)(uintptr_t)(g4out + i),
            (lv4i_p)(uint32_t)(uintptr_t)(l4 + i),
            0, 0);
    }
    wait_async0();   // S_ENDPGM also waits, but be explicit
#else
    for (int i = tid; i < nf4; i += TPB) g4out[i] = l4[i];
#endif
}

extern "C" void kernel_launch(void* const* d_in, const int* in_sizes, int n_in,
                              void* d_out, int out_size, void* d_ws, size_t ws_size,
                              hipStream_t stream) {
    const float* in  = (const float*)d_in[0];   // pred_pose, float32, N*132
    float*       out = (float*)d_out;           // float32, N*132
    (void)d_ws; (void)ws_size; (void)n_in;
    const int nSamples = in_sizes[0] / ROWF;    // 512*196 = 100352
    const int blocks   = (nSamples + TPB - 1) / TPB;  // 784
    pose6d_fk_kernel<<<blocks, TPB, 0, stream>>>(in, out, nSamples);
}